// ReBasedLinearAttention_30743375905047
// MI455X (gfx1250) — compile-verified
//
#include <hip/hip_runtime.h>
#include <hip/hip_bf16.h>

// ---------------------------------------------------------------------------
// ReBased linear attention (parallel/quadratic form), fp32 WMMA on gfx1250.
//   b=2, l=2048, hid=1024, H=16 heads, FD=16, HD=64.
//   All GEMMs use V_WMMA_F32_16X16X4_F32 (wave32, 16x16 C tile, K stepped by 4).
//   GEMM tile staging uses GLOBAL_LOAD_ASYNC_TO_LDS_B128 (ASYNCcnt-tracked)
//   when the toolchain exposes it; otherwise falls back to b128 load + ds store.
// ---------------------------------------------------------------------------

typedef __attribute__((ext_vector_type(2))) float v2f;
typedef __attribute__((ext_vector_type(4))) float v4f;
typedef __attribute__((ext_vector_type(8))) float v8f;
typedef __attribute__((ext_vector_type(4))) int   v4i;

typedef __attribute__((address_space(1))) v4i as1_v4i;   // global 16B vector
typedef __attribute__((address_space(3))) v4i as3_v4i;   // LDS 16B vector

#if defined(__has_builtin)
#if __has_builtin(__builtin_amdgcn_global_load_async_to_lds_b128) && \
    __has_builtin(__builtin_amdgcn_s_wait_asynccnt)
#define CDNA5_HAS_ASYNC_LDS 1
#endif
#endif

#define BB      2
#define LL      2048
#define HID     1024
#define NH      16
#define FD      16
#define HD      64
#define EPS_ATT 1e-5f
#define LN_EPS  1e-5f

static __device__ __forceinline__ v8f wmma_f32(v2f a, v2f b, v8f c) {
    // D = A(16x4) * B(4x16) + C(16x16), fp32.
    return __builtin_amdgcn_wmma_f32_16x16x4_f32(
        /*neg_a=*/false, a, /*neg_b=*/false, b,
        /*c_mod=*/(short)0, c, /*reuse_a=*/false, /*reuse_b=*/false);
}

// ---------------------------------------------------------------------------
// Generic C[M,N] = A[M,K] * W[N,K]^T   (M,N mult of 64; K mult of 16)
// Block: 256 threads = 8 waves; block tile 64x64; each wave owns a 32x16
// column strip (two 16x16 WMMA accumulators sharing one B operand).
// LDS tiles padded to stride 18 floats: keeps ds_load_b64 8B-aligned
// (even column offsets) while decorrelating bank indices across rows.
// Staging: async global->LDS (16B/lane), waited with s_wait_asynccnt 0
// before the consumer barrier.
// ---------------------------------------------------------------------------
__global__ __launch_bounds__(256) void gemm_xwT_wmma(
    const float* __restrict__ A,
    const float* __restrict__ W,
    float* __restrict__ C,
    int M, int N, int K)
{
    __shared__ float As[64][18];
    __shared__ float Ws[64][18];

    const int tid  = threadIdx.x;
    const int wave = tid >> 5;
    const int lane = tid & 31;
    const int lh   = lane >> 4;   // which half of the wave
    const int l16  = lane & 15;

    const int mBlock = blockIdx.y * 64;
    const int nBlock = blockIdx.x * 64;
    const int nSub   = (wave & 3) * 16;
    const int mSub   = (wave >> 2) * 32;

    v8f acc0 = {};
    v8f acc1 = {};

    const int ldRow = tid >> 2;        // 0..63
    const int ldCol = (tid & 3) * 4;   // 0,4,8,12

    for (int k0 = 0; k0 < K; k0 += 16) {
        const float* aSrc = &A[(size_t)(mBlock + ldRow) * K + k0 + ldCol];
        const float* wSrc = &W[(size_t)(nBlock + ldRow) * K + k0 + ldCol];

        __syncthreads();   // previous-iteration consumers done before overwrite
#if CDNA5_HAS_ASYNC_LDS
        // Direct HBM/L2 -> LDS copy, no VGPR round trip (ASYNCcnt-tracked).
        __builtin_amdgcn_global_load_async_to_lds_b128(
            (as1_v4i*)const_cast<float*>(aSrc),
            (as3_v4i*)&As[ldRow][ldCol], 0, 0);
        __builtin_amdgcn_global_load_async_to_lds_b128(
            (as1_v4i*)const_cast<float*>(wSrc),
            (as3_v4i*)&Ws[ldRow][ldCol], 0, 0);
        if (k0 + 16 < K) {
            __builtin_prefetch(aSrc + 16, 0, 3);
        }
        __builtin_amdgcn_s_wait_asynccnt(0);
#else
        *reinterpret_cast<v4f*>(&As[ldRow][ldCol]) = *reinterpret_cast<const v4f*>(aSrc);
        *reinterpret_cast<v4f*>(&Ws[ldRow][ldCol]) = *reinterpret_cast<const v4f*>(wSrc);
        if (k0 + 16 < K) {
            __builtin_prefetch(aSrc + 16, 0, 3);
        }
#endif
        __syncthreads();   // all waves' tiles visible

#pragma unroll
        for (int s = 0; s < 4; ++s) {
            const int kb = s * 4 + 2 * lh;  // even -> 8B aligned ds_load_b64
            v2f b  = *reinterpret_cast<const v2f*>(&Ws[nSub + l16][kb]);
            v2f a0 = *reinterpret_cast<const v2f*>(&As[mSub + l16][kb]);
            v2f a1 = *reinterpret_cast<const v2f*>(&As[mSub + 16 + l16][kb]);
            acc0 = wmma_f32(a0, b, acc0);
            acc1 = wmma_f32(a1, b, acc1);
        }
    }

    // C layout: VGPR r -> row (r + 8*half), lanes%16 -> column.
    const int col = nBlock + nSub + l16;
#pragma unroll
    for (int r = 0; r < 8; ++r) {
        const int row0 = mBlock + mSub + r + 8 * lh;
        C[(size_t)row0 * N + col]        = acc0[r];
        C[(size_t)(row0 + 16) * N + col] = acc1[r];
    }
}

// ---------------------------------------------------------------------------
// RebasedFeatureMap: per 16-vector layernorm (no affine) -> *gamma + beta,
// then uniform scale (FD^-0.5 for q, 1.0 for k). In-place.
// ---------------------------------------------------------------------------
__global__ __launch_bounds__(256) void feature_map_kernel(
    float* __restrict__ t,
    const float* __restrict__ gamma,
    const float* __restrict__ beta,
    int nvec, float scale)
{
    const int idx = blockIdx.x * blockDim.x + threadIdx.x;
    if (idx >= nvec) return;
    float* p = t + (size_t)idx * FD;

    float vals[FD];
#pragma unroll
    for (int i = 0; i < 4; ++i) {
        v4f q = *reinterpret_cast<const v4f*>(p + i * 4);
        vals[i * 4 + 0] = q[0]; vals[i * 4 + 1] = q[1];
        vals[i * 4 + 2] = q[2]; vals[i * 4 + 3] = q[3];
    }
    float mu = 0.f;
#pragma unroll
    for (int i = 0; i < FD; ++i) mu += vals[i];
    mu *= (1.0f / FD);
    float var = 0.f;
#pragma unroll
    for (int i = 0; i < FD; ++i) { float d = vals[i] - mu; var += d * d; }
    var *= (1.0f / FD);
    const float inv = rsqrtf(var + LN_EPS);
#pragma unroll
    for (int i = 0; i < FD; ++i)
        vals[i] = ((vals[i] - mu) * inv * gamma[i] + beta[i]) * scale;
#pragma unroll
    for (int i = 0; i < 4; ++i) {
        v4f q; q[0] = vals[i*4+0]; q[1] = vals[i*4+1]; q[2] = vals[i*4+2]; q[3] = vals[i*4+3];
        *reinterpret_cast<v4f*>(p + i * 4) = q;
    }
}

// ---------------------------------------------------------------------------
// Attention: s = (phi(q).phi(k))^2 masked causal; o = s@v / (rowsum(s)+eps).
// One wave owns one 16-query tile of one (b,h); streams key tiles jt<=qt.
//   S:    4x  WMMA (16x16x4)      -> C-layout registers
//   mask/square/z: elementwise in C layout
//   C->A relayout: bounce the 16x16 S tile through per-wave LDS
//   O+=SV: 16x WMMA per key tile  (N=64 as 4 chunks of 16)
// No block barriers (waves have different trip counts); same-wave DS
// ordering enforced with s_wait_dscnt 0.
// ---------------------------------------------------------------------------
__global__ __launch_bounds__(256) void rebased_attn_wmma(
    const float* __restrict__ q,   // (B, L, H*FD)
    const float* __restrict__ k,   // (B, L, H*FD)
    const float* __restrict__ v,   // (B, L, H*HD)
    float* __restrict__ o)         // (B, L, H*HD)
{
    __shared__ float Stile[8][16][18];   // per-wave 16x16 tile, padded stride

    const int wave = threadIdx.x >> 5;
    const int lane = threadIdx.x & 31;
    const int lh   = lane >> 4;
    const int l16  = lane & 15;

    const int bh = blockIdx.x;             // 0..B*H-1
    const int bb = bh / NH;
    const int hh = bh % NH;
    const int qt = blockIdx.y * 8 + wave;  // query tile 0..L/16-1

    // Cache phi(q) tile in A-layout registers: FD=16 -> 4 K-steps.
    v2f qa[4];
    {
        const float* qbase = q + ((size_t)bb * LL + (size_t)qt * 16 + l16) * (NH * FD) + hh * FD;
#pragma unroll
        for (int s = 0; s < 4; ++s) {
            const int kb = s * 4 + 2 * lh;
            qa[s] = *reinterpret_cast<const v2f*>(qbase + kb);
        }
    }

    v8f oacc[4] = {};
    v8f zacc = {};

    for (int jt = 0; jt <= qt; ++jt) {
        // ---- S = Q * K^T (16 queries x 16 keys, K=FD=16) ----
        v8f sacc = {};
        const float* kb_ptr = k + ((size_t)bb * LL + (size_t)jt * 16 + l16) * (NH * FD) + hh * FD;
#pragma unroll
        for (int s = 0; s < 4; ++s) {
            const int kb = s * 4 + 2 * lh;
            v2f bmat = *reinterpret_cast<const v2f*>(kb_ptr + kb);
            sacc = wmma_f32(qa[s], bmat, sacc);
        }

        // ---- square + causal mask + z rowsum; stage S to LDS ----
        const int iBase = qt * 16;
        const int jCol  = jt * 16 + l16;
#pragma unroll
        for (int r = 0; r < 8; ++r) {
            const int mRow = r + 8 * lh;
            float val = sacc[r];
            val = val * val;
            if (jCol > iBase + mRow) val = 0.0f;   // causal
            zacc[r] += val;
            Stile[wave][mRow][l16] = val;
        }
#if __has_builtin(__builtin_amdgcn_s_wait_dscnt)
        __builtin_amdgcn_s_wait_dscnt(0);          // same-wave LDS RAW fence
#endif

        // ---- O += S * Vtile : A = S (16x16), B = V rows (16 x 64) ----
        const float* vbase = v + ((size_t)bb * LL + (size_t)jt * 16) * (NH * HD) + hh * HD;
#pragma unroll
        for (int s = 0; s < 4; ++s) {
            const int kb = s * 4 + 2 * lh;
            v2f aS = *reinterpret_cast<const v2f*>(&Stile[wave][l16][kb]);
#pragma unroll
            for (int c = 0; c < 4; ++c) {
                v2f bV;
                bV[0] = vbase[(size_t)kb       * (NH * HD) + c * 16 + l16];
                bV[1] = vbase[(size_t)(kb + 1) * (NH * HD) + c * 16 + l16];
                oacc[c] = wmma_f32(aS, bV, oacc[c]);
            }
        }
    }

    // ---- z reduction: rows live in 16-lane halves; xor masks stay in-half ----
#pragma unroll
    for (int r = 0; r < 8; ++r) {
        float z = zacc[r];
        z += __shfl_xor(z, 1, 32);
        z += __shfl_xor(z, 2, 32);
        z += __shfl_xor(z, 4, 32);
        z += __shfl_xor(z, 8, 32);
        zacc[r] = z + EPS_ATT;
    }

    // ---- normalize + store (b, l, h*HD + d) ----
    float* obase = o + ((size_t)bb * LL + (size_t)qt * 16) * (NH * HD) + hh * HD;
#pragma unroll
    for (int r = 0; r < 8; ++r) {
        const int mRow = r + 8 * lh;
        const float invz = 1.0f / zacc[r];
#pragma unroll
        for (int c = 0; c < 4; ++c) {
            obase[(size_t)mRow * (NH * HD) + c * 16 + l16] = oacc[c][r] * invz;
        }
    }
}

// ---------------------------------------------------------------------------
// Launch: q = X@Wq^T ; k = X@Wk^T ; v = X@Wv^T ; feature-map q,k ;
//         attention ; out = O@Wo^T.  Workspace: 40 MiB of fp32 scratch.
// ---------------------------------------------------------------------------
extern "C" void kernel_launch(void* const* d_in, const int* in_sizes, int n_in,
                              void* d_out, int out_size, void* d_ws, size_t ws_size,
                              hipStream_t stream) {
    const float* x     = (const float*)d_in[0];
    const float* Wq    = (const float*)d_in[1];
    const float* Wk    = (const float*)d_in[2];
    const float* Wv    = (const float*)d_in[3];
    const float* Wo    = (const float*)d_in[4];
    const float* gamma = (const float*)d_in[5];
    const float* beta  = (const float*)d_in[6];
    float* out = (float*)d_out;

    const int M = BB * LL;            // 4096 rows
    float* ws   = (float*)d_ws;
    float* qbuf = ws;                                  // 4096*256
    float* kbuf = qbuf + (size_t)M * (NH * FD);        // 4096*256
    float* vbuf = kbuf + (size_t)M * (NH * FD);        // 4096*1024
    float* obuf = vbuf + (size_t)M * HID;              // 4096*1024

    dim3 blk(256);

    // Projections (fp32 WMMA GEMMs)
    gemm_xwT_wmma<<<dim3((NH * FD) / 64, M / 64), blk, 0, stream>>>(x, Wq, qbuf, M, NH * FD, HID);
    gemm_xwT_wmma<<<dim3((NH * FD) / 64, M / 64), blk, 0, stream>>>(x, Wk, kbuf, M, NH * FD, HID);
    gemm_xwT_wmma<<<dim3(HID / 64,      M / 64), blk, 0, stream>>>(x, Wv, vbuf, M, HID, HID);

    // Feature map (layernorm -> gamma/beta); q also scaled by FD^-0.5
    const int nvec = M * NH;          // 65536 16-vectors
    feature_map_kernel<<<dim3((nvec + 255) / 256), blk, 0, stream>>>(qbuf, gamma, beta, nvec, 0.25f);
    feature_map_kernel<<<dim3((nvec + 255) / 256), blk, 0, stream>>>(kbuf, gamma, beta, nvec, 1.0f);

    // Causal squared linear attention (WMMA), normalized
    rebased_attn_wmma<<<dim3(BB * NH, (LL / 16) / 8), blk, 0, stream>>>(qbuf, kbuf, vbuf, obuf);

    // Output projection
    gemm_xwT_wmma<<<dim3(HID / 64, M / 64), blk, 0, stream>>>(obuf, Wo, out, M, HID, HID);
}